// ScaledDotAttention_46050639348197
// MI455X (gfx1250) — compile-verified
//
#include <hip/hip_runtime.h>
#include <stdint.h>

typedef __attribute__((ext_vector_type(2))) float v2f;
typedef __attribute__((ext_vector_type(8))) float v8f;

#define MASKED_VAL (-2147483648.0f)

#define TM 64
#define TN 64
#define TK 32
#define A_PAD 36   // 144B row stride: 16B aligned, period-16 bank pattern
#define B_PAD 68   // 272B row stride: 16B aligned

#define ASYNC_WAIT() asm volatile("s_wait_asynccnt 0x0" ::: "memory")

// ---------------------------------------------------------------------------
// Async global->LDS copy of one 16B chunk per call (CDNA5 ASYNCcnt path).
// LDS offset = low 32 bits of the generic shared-memory address (ISA 10.2:
// LDS aperture address truncates to the 0-based LDS offset).
// ---------------------------------------------------------------------------
__device__ inline void async_copy_b128(const float* g, const float* lds_ptr) {
  unsigned lds = (unsigned)(uintptr_t)lds_ptr;
  asm volatile("global_load_async_to_lds_b128 %0, %1, off"
               :
               : "v"(lds), "v"(g)
               : "memory");
}

// 64 rows x 32 cols row-major tile fill (A tiles, and B tiles stored [n][k]).
__device__ inline void fill_tile64x32_async(float (*tile)[A_PAD],
                                            const float* __restrict__ g,
                                            int ld, int row0, int col0,
                                            int tid) {
#pragma unroll
  for (int j = 0; j < 2; ++j) {
    int fidx = tid + j * 256;       // 0..511 chunks of 16B
    int r = fidx >> 3;
    int c4 = (fidx & 7) << 2;
    async_copy_b128(g + (size_t)(row0 + r) * ld + col0 + c4, &tile[r][c4]);
  }
}

// 32 rows x 64 cols row-major tile fill (Vp tile stored [k][n]).
__device__ inline void fill_tile32x64_async(float (*tile)[B_PAD],
                                            const float* __restrict__ g,
                                            int ld, int row0, int col0,
                                            int tid) {
#pragma unroll
  for (int j = 0; j < 2; ++j) {
    int fidx = tid + j * 256;
    int k = fidx >> 4;
    int c4 = (fidx & 15) << 2;
    async_copy_b128(g + (size_t)(row0 + k) * ld + col0 + c4, &tile[k][c4]);
  }
}

// ---------------------------------------------------------------------------
// WMMA f32 16x16x4 inner loops. Wave layout: mi = wave&3 picks 16-row strip,
// nj = (wave>>2)*2 picks two 16-col subtiles. Per-lane fragments:
//   A: m = mi*16 + lane%16, holds A[m][kb], A[m][kb+1], kb = 2*(lane/16)
//   B: n = nj*16 + lane%16, holds B[kb][n], B[kb+1][n]
// ---------------------------------------------------------------------------
__device__ inline void mma_nt(const float (*As)[A_PAD], const float (*Bn)[A_PAD],
                              int mrow, int nc0, int nc1, int kb,
                              v8f& acc0, v8f& acc1) {
#pragma unroll
  for (int kk = 0; kk < TK; kk += 4) {
    v2f a, b0, b1;
    a.x = As[mrow][kk + kb];
    a.y = As[mrow][kk + kb + 1];
    b0.x = Bn[nc0][kk + kb];
    b0.y = Bn[nc0][kk + kb + 1];
    b1.x = Bn[nc1][kk + kb];
    b1.y = Bn[nc1][kk + kb + 1];
    acc0 = __builtin_amdgcn_wmma_f32_16x16x4_f32(false, a, false, b0,
                                                 (short)0, acc0, false, false);
    acc1 = __builtin_amdgcn_wmma_f32_16x16x4_f32(false, a, false, b1,
                                                 (short)0, acc1, false, false);
  }
}

__device__ inline void mma_nn(const float (*As)[A_PAD], const float (*Bk)[B_PAD],
                              int mrow, int nc0, int nc1, int kb,
                              v8f& acc0, v8f& acc1) {
#pragma unroll
  for (int kk = 0; kk < TK; kk += 4) {
    v2f a, b0, b1;
    a.x = As[mrow][kk + kb];
    a.y = As[mrow][kk + kb + 1];
    b0.x = Bk[kk + kb][nc0];
    b0.y = Bk[kk + kb + 1][nc0];
    b1.x = Bk[kk + kb][nc1];
    b1.y = Bk[kk + kb + 1][nc1];
    acc0 = __builtin_amdgcn_wmma_f32_16x16x4_f32(false, a, false, b0,
                                                 (short)0, acc0, false, false);
    acc1 = __builtin_amdgcn_wmma_f32_16x16x4_f32(false, a, false, b1,
                                                 (short)0, acc1, false, false);
  }
}

// ---------------------------------------------------------------------------
// Stage 1: Y[m][n] = sum_k X[m][k] * W[n][k] + bias[n]   (torch Linear, NT)
// Double-buffered async LDS fills overlapped with the WMMA loop.
// ---------------------------------------------------------------------------
__global__ __launch_bounds__(256)
void proj_kernel(const float* __restrict__ X, const float* __restrict__ W,
                 const float* __restrict__ bias, float* __restrict__ Y,
                 int N, int K) {
  __shared__ float As[2][TM][A_PAD];
  __shared__ float Bn[2][TN][A_PAD];
  const int tid = threadIdx.x;
  const int lane = tid & 31;
  const int wave = tid >> 5;
  const int mi = wave & 3;
  const int nj = (wave >> 2) << 1;
  const int m0 = blockIdx.x * TM;
  const int n0 = blockIdx.y * TN;
  const int lm = lane & 15;
  const int kb = (lane >> 4) << 1;
  const int mrow = mi * 16 + lm;
  const int nc0 = nj * 16 + lm;
  const int nc1 = nc0 + 16;

  v8f acc0 = {};
  v8f acc1 = {};

  fill_tile64x32_async(As[0], X, K, m0, 0, tid);
  fill_tile64x32_async(Bn[0], W, K, n0, 0, tid);
  ASYNC_WAIT();
  __syncthreads();

  int cur = 0;
  for (int kt = 0; kt < K; kt += TK) {
    const int nxt = cur ^ 1;
    if (kt + TK < K) {
      fill_tile64x32_async(As[nxt], X, K, m0, kt + TK, tid);
      fill_tile64x32_async(Bn[nxt], W, K, n0, kt + TK, tid);
    }
    mma_nt(As[cur], Bn[cur], mrow, nc0, nc1, kb, acc0, acc1);
    ASYNC_WAIT();
    __syncthreads();
    cur = nxt;
  }

  const float bn0 = bias[n0 + nc0];
  const float bn1 = bias[n0 + nc1];
  const int mbase = m0 + mi * 16 + ((lane >> 4) << 3);
#pragma unroll
  for (int i = 0; i < 8; ++i) {
    size_t row = (size_t)(mbase + i);
    Y[row * N + n0 + nc0] = acc0[i] + bn0;
    Y[row * N + n0 + nc1] = acc1[i] + bn1;
  }
}

// ---------------------------------------------------------------------------
// Stage 2: E[b][q][k] = dot(Qp[b][q], Kp[b][k]); masked keys -> -2^31
// ---------------------------------------------------------------------------
__global__ __launch_bounds__(256)
void scores_kernel(const float* __restrict__ Qp, const float* __restrict__ Kp,
                   const int* __restrict__ kmask, float* __restrict__ E,
                   int L, int D) {
  __shared__ float As[2][TM][A_PAD];
  __shared__ float Bn[2][TN][A_PAD];
  const int b = blockIdx.z;
  const float* A = Qp + (size_t)b * L * D;
  const float* Bm = Kp + (size_t)b * L * D;
  float* Eb = E + (size_t)b * L * L;

  const int tid = threadIdx.x;
  const int lane = tid & 31;
  const int wave = tid >> 5;
  const int mi = wave & 3;
  const int nj = (wave >> 2) << 1;
  const int m0 = blockIdx.x * TM;   // query tile
  const int n0 = blockIdx.y * TN;   // key tile
  const int lm = lane & 15;
  const int kb = (lane >> 4) << 1;
  const int mrow = mi * 16 + lm;
  const int nc0 = nj * 16 + lm;
  const int nc1 = nc0 + 16;

  v8f acc0 = {};
  v8f acc1 = {};

  fill_tile64x32_async(As[0], A, D, m0, 0, tid);
  fill_tile64x32_async(Bn[0], Bm, D, n0, 0, tid);
  ASYNC_WAIT();
  __syncthreads();

  int cur = 0;
  for (int kt = 0; kt < D; kt += TK) {
    const int nxt = cur ^ 1;
    if (kt + TK < D) {
      fill_tile64x32_async(As[nxt], A, D, m0, kt + TK, tid);
      fill_tile64x32_async(Bn[nxt], Bm, D, n0, kt + TK, tid);
    }
    mma_nt(As[cur], Bn[cur], mrow, nc0, nc1, kb, acc0, acc1);
    ASYNC_WAIT();
    __syncthreads();
    cur = nxt;
  }

  const int kcol0 = n0 + nc0;
  const int kcol1 = n0 + nc1;
  const int km0 = kmask[(size_t)b * L + kcol0];
  const int km1 = kmask[(size_t)b * L + kcol1];
  const int mbase = m0 + mi * 16 + ((lane >> 4) << 3);
#pragma unroll
  for (int i = 0; i < 8; ++i) {
    size_t row = (size_t)(mbase + i);
    Eb[row * L + kcol0] = km0 ? acc0[i] : MASKED_VAL;
    Eb[row * L + kcol1] = km1 ? acc1[i] : MASKED_VAL;
  }
}

// ---------------------------------------------------------------------------
// Stage 3a: per-row max and sum(exp(x-max)); one wave32 per row.
// ---------------------------------------------------------------------------
__global__ __launch_bounds__(256)
void rowstats_kernel(const float* __restrict__ E, float* __restrict__ rmax,
                     float* __restrict__ rsum, int Lk) {
  const int wave = threadIdx.x >> 5;
  const int lane = threadIdx.x & 31;
  const size_t row = (size_t)blockIdx.x * 8 + wave;
  const float4* e4 = (const float4*)(E + row * Lk);
  const int n4 = Lk >> 2;

  float m = -3.4e38f;
  for (int i = lane; i < n4; i += 32) {
    float4 v = e4[i];
    m = fmaxf(m, fmaxf(fmaxf(v.x, v.y), fmaxf(v.z, v.w)));
  }
#pragma unroll
  for (int off = 16; off; off >>= 1) m = fmaxf(m, __shfl_xor(m, off, 32));

  float s = 0.0f;
  for (int i = lane; i < n4; i += 32) {
    float4 v = e4[i];
    s += __expf(v.x - m) + __expf(v.y - m) + __expf(v.z - m) + __expf(v.w - m);
  }
#pragma unroll
  for (int off = 16; off; off >>= 1) s += __shfl_xor(s, off, 32);

  if (lane == 0) {
    rmax[row] = m;
    rsum[row] = s;
  }
}

// ---------------------------------------------------------------------------
// Stage 3b: Out[b][q][d] = (exp(E-rowmax) @ Vp) / rowsum, zeroed by q_mask.
// Vp tile: async LDS fill. P tile: register-staged pipeline (loads for the
// next chunk issued before the WMMA loop, exp + ds_store after it).
// ---------------------------------------------------------------------------
__global__ __launch_bounds__(256)
void attnout_kernel(const float* __restrict__ E, const float* __restrict__ rmax,
                    const float* __restrict__ rsum, const float* __restrict__ Vp,
                    const int* __restrict__ qmask, float* __restrict__ Out,
                    int L, int D) {
  __shared__ float As[2][TM][A_PAD];
  __shared__ float Bk[2][TK][B_PAD];
  const int b = blockIdx.z;
  const float* Eb = E + (size_t)b * L * L;
  const float* Vb = Vp + (size_t)b * L * D;

  const int tid = threadIdx.x;
  const int lane = tid & 31;
  const int wave = tid >> 5;
  const int mi = wave & 3;
  const int nj = (wave >> 2) << 1;
  const int q0 = blockIdx.x * TM;
  const int d0 = blockIdx.y * TN;
  const int lm = lane & 15;
  const int kb = (lane >> 4) << 1;
  const int mrow = mi * 16 + lm;
  const int nc0 = nj * 16 + lm;
  const int nc1 = nc0 + 16;

  // Loop-invariant A-fill coordinates and rowmax values.
  int fr[2], fc[2];
  float fmx[2];
#pragma unroll
  for (int j = 0; j < 2; ++j) {
    int fidx = tid + j * 256;
    fr[j] = fidx >> 3;
    fc[j] = (fidx & 7) << 2;
    fmx[j] = rmax[(size_t)b * L + q0 + fr[j]];
  }

  v8f acc0 = {};
  v8f acc1 = {};

  // Prologue: chunk 0.
  fill_tile32x64_async(Bk[0], Vb, D, 0, d0, tid);
  {
    float4 va[2];
#pragma unroll
    for (int j = 0; j < 2; ++j)
      va[j] = *(const float4*)(Eb + (size_t)(q0 + fr[j]) * L + fc[j]);
#pragma unroll
    for (int j = 0; j < 2; ++j) {
      float4 p;
      p.x = __expf(va[j].x - fmx[j]);
      p.y = __expf(va[j].y - fmx[j]);
      p.z = __expf(va[j].z - fmx[j]);
      p.w = __expf(va[j].w - fmx[j]);
      *(float4*)&As[0][fr[j]][fc[j]] = p;
    }
  }
  ASYNC_WAIT();
  __syncthreads();

  int cur = 0;
  for (int kt = 0; kt < L; kt += TK) {
    const int nxt = cur ^ 1;
    const bool more = (kt + TK) < L;
    float4 va[2];
    if (more) {
      fill_tile32x64_async(Bk[nxt], Vb, D, kt + TK, d0, tid);
#pragma unroll
      for (int j = 0; j < 2; ++j)
        va[j] = *(const float4*)(Eb + (size_t)(q0 + fr[j]) * L + kt + TK + fc[j]);
    }
    mma_nn(As[cur], Bk[cur], mrow, nc0, nc1, kb, acc0, acc1);
    if (more) {
#pragma unroll
      for (int j = 0; j < 2; ++j) {
        float4 p;
        p.x = __expf(va[j].x - fmx[j]);
        p.y = __expf(va[j].y - fmx[j]);
        p.z = __expf(va[j].z - fmx[j]);
        p.w = __expf(va[j].w - fmx[j]);
        *(float4*)&As[nxt][fr[j]][fc[j]] = p;
      }
    }
    ASYNC_WAIT();
    __syncthreads();
    cur = nxt;
  }

  const int mbase = q0 + mi * 16 + ((lane >> 4) << 3);
#pragma unroll
  for (int i = 0; i < 8; ++i) {
    int r = mbase + i;
    float scale = qmask[(size_t)b * L + r] ? (1.0f / rsum[(size_t)b * L + r])
                                           : 0.0f;
    size_t o = ((size_t)b * L + r) * D;
    Out[o + d0 + nc0] = acc0[i] * scale;
    Out[o + d0 + nc1] = acc1[i] * scale;
  }
}

// ---------------------------------------------------------------------------
extern "C" void kernel_launch(void* const* d_in, const int* in_sizes, int n_in,
                              void* d_out, int out_size, void* d_ws,
                              size_t ws_size, hipStream_t stream) {
  const float* Q  = (const float*)d_in[0];
  const float* K  = (const float*)d_in[1];
  const float* V  = (const float*)d_in[2];
  const float* Wq = (const float*)d_in[3];
  const float* bq = (const float*)d_in[4];
  const float* Wk = (const float*)d_in[5];
  const float* bk = (const float*)d_in[6];
  const float* Wv = (const float*)d_in[7];
  const float* bv = (const float*)d_in[8];
  const int* qm   = (const int*)d_in[9];
  const int* km   = (const int*)d_in[10];

  const int B = 8, L = 2048, D = 1024;
  const int M = B * L;

  float* ws   = (float*)d_ws;
  float* Qp   = ws;
  float* Kp   = Qp + (size_t)M * D;
  float* Vp   = Kp + (size_t)M * D;
  float* rmax = Vp + (size_t)M * D;
  float* rsum = rmax + M;

  float* Aout = (float*)d_out;                 // [B, L, D]
  float* Eout = Aout + (size_t)M * D;          // [B, L, L]

  dim3 blk(256, 1, 1);

  dim3 gp(M / TM, D / TN, 1);
  proj_kernel<<<gp, blk, 0, stream>>>(Q, Wq, bq, Qp, D, D);
  proj_kernel<<<gp, blk, 0, stream>>>(K, Wk, bk, Kp, D, D);
  proj_kernel<<<gp, blk, 0, stream>>>(V, Wv, bv, Vp, D, D);

  dim3 gs(L / TM, L / TN, B);
  scores_kernel<<<gs, blk, 0, stream>>>(Qp, Kp, km, Eout, L, D);

  rowstats_kernel<<<dim3(M / 8, 1, 1), blk, 0, stream>>>(Eout, rmax, rsum, L);

  dim3 ga(L / TM, D / TN, B);
  attnout_kernel<<<ga, blk, 0, stream>>>(Eout, rmax, rsum, Vp, qm, Aout, L, D);
}